// cal_sf_by_net_59047210385783
// MI455X (gfx1250) — compile-verified
//
#include <hip/hip_runtime.h>
#include <math.h>

// Problem constants (reference: x[4,32,1024,1024] f32, radius = W/40 = 25)
#define BATCH 4
#define CHAN  32
#define H     1024
#define W     1024
#define STRIP 8                                  // rows per block in stage 1

constexpr int R    = W / 40;                     // 25
constexpr int KLO  = (R + 3) & ~3;               // 28  (aligned halo below)
constexpr int KHI  = ((15 + R) & ~3) + 3;        // 43  (halo above, incl. K-block tail)
constexpr int NKB  = (KLO + KHI + 1) / 4;        // 18 K-blocks of 4 per 16-wide tile

typedef __attribute__((ext_vector_type(2))) float v2f;
typedef __attribute__((ext_vector_type(8))) float v8f;

// ---------------------------------------------------------------------------
// Stage 1: g[b,h,w] = sum_c sqrt((x[b,c,h,w-1]-x)^2 + (x[b,c,h-1,w]-x)^2)
// Streaming: 512 MB read (dominant cost, ~22us at 23.3 TB/s), 16 MB write.
// 8-row strips keep the h-1 row in registers -> halo overfetch only 9/8.
// ---------------------------------------------------------------------------
__global__ __launch_bounds__(256)
void grad_chansum_kernel(const float* __restrict__ x, float* __restrict__ g) {
    const int tid = threadIdx.x;                 // 0..255 covers one full row
    const int w0  = tid * 4;
    const int strip = blockIdx.x;
    const int b  = strip / (H / STRIP);
    const int h0 = (strip % (H / STRIP)) * STRIP;

    const float* xb = x + (size_t)b * CHAN * H * W;

    float acc[STRIP][4];
    #pragma unroll
    for (int r = 0; r < STRIP; ++r)
        acc[r][0] = acc[r][1] = acc[r][2] = acc[r][3] = 0.f;

    for (int c = 0; c < CHAN; ++c) {
        const float* plane = xb + (size_t)c * H * W;
        if (c + 1 < CHAN)   // speculative prefetch of next channel's strip head
            __builtin_prefetch(plane + (size_t)H * W + (size_t)h0 * W + w0, 0, 1);

        float4 prev;
        if (h0 == 0) prev = make_float4(0.f, 0.f, 0.f, 0.f);
        else         prev = *(const float4*)(plane + (size_t)(h0 - 1) * W + w0);

        #pragma unroll
        for (int r = 0; r < STRIP; ++r) {
            const float* rowp = plane + (size_t)(h0 + r) * W;
            float4 cur = *(const float4*)(rowp + w0);
            float left = (w0 == 0) ? 0.f : rowp[w0 - 1];   // same cacheline
            float lv[4] = {left,   cur.x, cur.y, cur.z};
            float cv[4] = {cur.x,  cur.y, cur.z, cur.w};
            float uv[4] = {prev.x, prev.y, prev.z, prev.w};
            #pragma unroll
            for (int j = 0; j < 4; ++j) {
                float dx = lv[j] - cv[j];
                float dy = uv[j] - cv[j];
                acc[r][j] += sqrtf(dx * dx + dy * dy);
            }
            prev = cur;
        }
    }

    float* gout = g + ((size_t)b * H + h0) * W + w0;
    #pragma unroll
    for (int r = 0; r < STRIP; ++r)
        *(float4*)(gout + (size_t)r * W) =
            make_float4(acc[r][0], acc[r][1], acc[r][2], acc[r][3]);
}

// ---------------------------------------------------------------------------
// Stage 2: box sum along w as banded-ones matmul on the WMMA pipe.
// Block = 8 waves = 16 rows x 128 output cols. The 16 x 184 halo strip is
// cooperatively staged (coalesced, zero-padded) into LDS once; each wave then
// runs 18 fully-unrolled V_WMMA_F32_16X16X4_F32 with A fed by ds_load_b64 and
// B the band indicator built branchlessly in registers (EXEC stays all-1s).
// A layout (16x4 f32): lanes 0-15 row M=lane hold K0,K1; lanes 16-31 K2,K3.
// B layout (4x16 f32): lanes 0-15 col N=lane hold K0,K1; lanes 16-31 K2,K3.
// ---------------------------------------------------------------------------
constexpr int LDS_W      = 112 + KLO + KHI + 1;  // 184 cols staged per block
constexpr int LDS_STRIDE = LDS_W + 2;            // 186: even (8B align) + bank spread

__global__ __launch_bounds__(256)
void box_w_wmma_kernel(const float* __restrict__ g, float* __restrict__ t) {
    __shared__ __align__(16) float smem[16 * LDS_STRIDE];

    const int lane = threadIdx.x & 31;
    const int wave = __builtin_amdgcn_readfirstlane(threadIdx.x >> 5);
    const int blk  = __builtin_amdgcn_readfirstlane(blockIdx.x);

    const int tilesY = H / 16, groupsX = W / 128;
    const int b   = blk / (tilesY * groupsX);
    const int rem = blk % (tilesY * groupsX);
    const int r0  = (rem / groupsX) * 16;
    const int xg  = (rem % groupsX) * 128;       // first output col of block

    const float* img = g + ((size_t)b * H + r0) * W;

    // Cooperative coalesced fill (zero padding applied here, once).
    for (int tI = threadIdx.x; tI < 16 * LDS_W; tI += 256) {
        const int rr = tI / LDS_W;
        const int cc = tI % LDS_W;
        const int gc = xg - KLO + cc;
        const int gcc = min(max(gc, 0), W - 1);          // clamped: load always valid
        const float v = img[(size_t)rr * W + gcc];
        smem[rr * LDS_STRIDE + cc] = ((unsigned)gc < (unsigned)W) ? v : 0.f;
    }
    __syncthreads();

    const int half = lane >> 4;                  // 0 -> K0/K1, 1 -> K2/K3
    const int l15  = lane & 15;
    const float* srow = smem + l15 * LDS_STRIDE;
    const int cbase = wave * 16 + half * 2;      // LDS col of rel = -KLO

    v8f acc = {0.f, 0.f, 0.f, 0.f, 0.f, 0.f, 0.f, 0.f};

    #pragma unroll
    for (int i = 0; i < NKB; ++i) {
        const int rel = -KLO + 4 * i + half * 2; // k offset from tile col x0
        const float2 av = *(const float2*)(srow + cbase + 4 * i);   // 8B aligned
        v2f a;  a.x = av.x;  a.y = av.y;
        v2f bm;                                  // band: |k - n| <= R, lane-static
        bm.x = (rel     >= l15 - R && rel     <= l15 + R) ? 1.f : 0.f;
        bm.y = (rel + 1 >= l15 - R && rel + 1 <= l15 + R) ? 1.f : 0.f;
        acc = __builtin_amdgcn_wmma_f32_16x16x4_f32(
            false, a, false, bm, (short)0, acc, false, false);
    }

    // C/D layout: VGPR i -> M=i (lanes 0-15) / M=8+i (lanes 16-31), N=lane&15
    float* out = t + ((size_t)b * H + r0) * W + xg + wave * 16 + l15;
    #pragma unroll
    for (int i = 0; i < 8; ++i) {
        const int m = half ? (8 + i) : i;
        out[(size_t)m * W] = acc[i];
    }
}

// ---------------------------------------------------------------------------
// Stage 3: box sum along h. D = A(band 16xK) x B(K x 16 data tile).
// B loads are 16 consecutive floats per half-wave (64B, coalesced); row index
// is clamped + value-masked so there is no exec-mask branching around WMMA.
// ---------------------------------------------------------------------------
__global__ __launch_bounds__(256)
void box_h_wmma_kernel(const float* __restrict__ t, float* __restrict__ out) {
    const int lane = threadIdx.x & 31;
    const int tile = __builtin_amdgcn_readfirstlane(blockIdx.x * 8 + (threadIdx.x >> 5));

    const int tilesPerImg = (H / 16) * (W / 16);
    const int b   = tile / tilesPerImg;
    const int rem = tile % tilesPerImg;
    const int r0  = (rem / (W / 16)) * 16;
    const int x0  = (rem % (W / 16)) * 16;

    const int half = lane >> 4;
    const int l15  = lane & 15;
    const float* img = t + (size_t)b * H * W + x0 + l15;

    v8f acc = {0.f, 0.f, 0.f, 0.f, 0.f, 0.f, 0.f, 0.f};

    #pragma unroll
    for (int i = 0; i < NKB; ++i) {
        const int rel  = -KLO + 4 * i + half * 2;        // row offset from r0
        const int row0 = r0 + rel;
        const int row1 = row0 + 1;
        const int rc0  = min(max(row0, 0), H - 1);       // clamped: always valid
        const int rc1  = min(max(row1, 0), H - 1);
        const float b0 = img[(size_t)rc0 * W];
        const float b1 = img[(size_t)rc1 * W];
        v2f bm;
        bm.x = ((unsigned)row0 < (unsigned)H) ? b0 : 0.f;
        bm.y = ((unsigned)row1 < (unsigned)H) ? b1 : 0.f;
        v2f a;                                           // band: |rel - m| <= R
        a.x = (rel     >= l15 - R && rel     <= l15 + R) ? 1.f : 0.f;
        a.y = (rel + 1 >= l15 - R && rel + 1 <= l15 + R) ? 1.f : 0.f;
        acc = __builtin_amdgcn_wmma_f32_16x16x4_f32(
            false, a, false, bm, (short)0, acc, false, false);
    }

    float* o = out + (size_t)b * H * W + (size_t)r0 * W + x0 + l15;
    #pragma unroll
    for (int i = 0; i < 8; ++i) {
        const int m = half ? (8 + i) : i;
        o[(size_t)m * W] = acc[i];
    }
}

// ---------------------------------------------------------------------------
extern "C" void kernel_launch(void* const* d_in, const int* in_sizes, int n_in,
                              void* d_out, int out_size, void* d_ws, size_t ws_size,
                              hipStream_t stream) {
    const float* x = (const float*)d_in[0];      // [4,32,1024,1024] f32
    float* out = (float*)d_out;                  // [4,1024,1024]   f32
    float* tmp = (float*)d_ws;                   // 16 MB scratch

    // Stage 1: gradient magnitude + channel sum -> d_out (temp for g)
    grad_chansum_kernel<<<BATCH * (H / STRIP), 256, 0, stream>>>(x, out);

    // Stage 2: box along w (WMMA + LDS staging): d_out -> d_ws
    const int blocks2 = BATCH * (H / 16) * (W / 128); // 2048
    box_w_wmma_kernel<<<blocks2, 256, 0, stream>>>(out, tmp);

    // Stage 3: box along h (WMMA): d_ws -> d_out
    const int blocks3 = BATCH * (H / 16) * (W / 16) / 8; // 2048
    box_h_wmma_kernel<<<blocks3, 256, 0, stream>>>(tmp, out);
}